// _GCNMRConv2d_42554535969008
// MI455X (gfx1250) — compile-verified
//
#include <hip/hip_runtime.h>
#include <hip/hip_bf16.h>
#include <math.h>

// Problem constants (from reference)
#define BB    2
#define CC    192
#define NN    16384
#define KK    16
#define CO    192
#define CIN2  384          // 2*C
#define EPSBN 1e-5f

typedef __attribute__((ext_vector_type(2))) float v2f;
typedef __attribute__((ext_vector_type(8))) float v8f;
typedef __attribute__((ext_vector_type(4))) int   v4i;

// --------------------------------------------------------------------------
// CDNA5 async global->LDS copy (ASYNCcnt path), with safe fallback.
// Probe round 2 confirmed __builtin_amdgcn_global_load_async_to_lds_b128
// exists and takes int4-vector pointers.
// --------------------------------------------------------------------------
#if defined(__has_builtin)
#if __has_builtin(__builtin_amdgcn_global_load_async_to_lds_b128)
#define HAVE_ASYNC_LDS 1
#endif
#endif

__device__ __forceinline__ void wait_async_zero() {
#if defined(__has_builtin) && __has_builtin(__builtin_amdgcn_s_wait_asynccnt)
  __builtin_amdgcn_s_wait_asynccnt(0);
#else
  asm volatile("s_wait_asynccnt 0x0" ::: "memory");
#endif
}

// Copy one 64KB channel row (NN floats) global -> LDS, 256 threads.
__device__ __forceinline__ void row_copy_async(const float* __restrict__ g,
                                               float* l) {
#if defined(HAVE_ASYNC_LDS)
  // 16 x b128 per thread, fully coalesced; tracked by ASYNCcnt.
#pragma unroll
  for (int it = 0; it < (NN / 4) / 256; ++it) {
    const int off = (it * 256 + (int)threadIdx.x) * 4;   // float index, 16B aligned
    __builtin_amdgcn_global_load_async_to_lds_b128(
        (v4i*)(g + off), (v4i*)(l + off),
        /*imm offset*/ 0, /*cpol*/ 0);
  }
#else
  const float4* g4 = (const float4*)g;
  float4* l4 = (float4*)l;
#pragma unroll
  for (int it = 0; it < (NN / 4) / 256; ++it)
    l4[it * 256 + threadIdx.x] = g4[it * 256 + threadIdx.x];
#endif
}

// ---------------------------------------------------------------------------
// Kernel 1: gather + max-relative feature.
//   mr[b,c,n] = max_k ( x[b,c, idx_j[b,n,k]] - x[b,c, idx_i[b,n,k]] )
// Block owns (batch b, 12-channel group, 256-node tile).
//   - each thread keeps its node's 32 indices in VGPRs (channel-invariant,
//     contiguous 64B+64B global reads, loaded once)
//   - channel rows double-buffered in LDS (2 x 64KB) with async global->LDS
//     prefetch of row c+1 overlapping the 8192 LDS gathers on row c.
// ---------------------------------------------------------------------------
#define TILE_N 256
#define CGROUP 12

__global__ __launch_bounds__(256)
void k_gather_mr(const float* __restrict__ x,
                 const int*   __restrict__ eidx,   // [2, B, N, K]
                 float*       __restrict__ mr)     // [B, C, N]
{
  extern __shared__ char smem[];
  float* buf0 = (float*)smem;                       // NN floats
  float* buf1 = (float*)(smem + NN * sizeof(float)); // NN floats

  const int b  = blockIdx.z;
  const int c0 = blockIdx.y * CGROUP;
  const int n0 = blockIdx.x * TILE_N;
  const int n  = threadIdx.x;                       // one node per thread

  // edge_index offset(e,b,n,k) = ((e*B + b)*N + n)*K + k ; K==16 ints == 64B
  const int4* pj4 = (const int4*)(eidx + ((0 * BB + b) * NN + n0 + n) * KK);
  const int4* pi4 = (const int4*)(eidx + ((1 * BB + b) * NN + n0 + n) * KK);
  int jv[KK], iv[KK];
#pragma unroll
  for (int q = 0; q < 4; ++q) {
    const int4 j = pj4[q], i = pi4[q];
    jv[q * 4 + 0] = j.x; jv[q * 4 + 1] = j.y; jv[q * 4 + 2] = j.z; jv[q * 4 + 3] = j.w;
    iv[q * 4 + 0] = i.x; iv[q * 4 + 1] = i.y; iv[q * 4 + 2] = i.z; iv[q * 4 + 3] = i.w;
  }

  const float* xb = x + (size_t)b * CC * NN;

  // prefetch first row
  row_copy_async(xb + (size_t)c0 * NN, buf0);

  for (int cc = 0; cc < CGROUP; ++cc) {
    wait_async_zero();       // this wave's slice of current buffer landed
    __syncthreads();         // whole buffer visible; prev buffer's readers done
    if (cc + 1 < CGROUP) {
      float* nxt = ((cc + 1) & 1) ? buf1 : buf0;
      row_copy_async(xb + (size_t)(c0 + cc + 1) * NN, nxt);
    }
    const float* s = (cc & 1) ? buf1 : buf0;

    float m = -INFINITY;
#pragma unroll
    for (int k = 0; k < KK; ++k) {
      m = fmaxf(m, s[jv[k]] - s[iv[k]]);
    }
    mr[((size_t)b * CC + (c0 + cc)) * NN + n0 + n] = m;
  }
}

// ---------------------------------------------------------------------------
// Kernel 2: 1x1 conv as WMMA GEMM.
//   h[b,o,n] = sum_{c2=0..383} W[o,c2] * cat[b,c2,n] + b_conv[o]
//   cat[b,2c,n] = x[b,c,n] ; cat[b,2c+1,n] = mr[b,c,n]   (never materialized)
// One wave computes one 16(o) x 16(n) tile via 96 steps of
// V_WMMA_F32_16X16X4_F32.  W row-block staged in LDS (padded stride).
// A-frag layout (16x4 f32): lanes 0-15 -> K={0,1}, lanes 16-31 -> K={2,3}.
// B-frag: b.x = even c2 (from x), b.y = odd c2 (from mr).
// ---------------------------------------------------------------------------
#define SWSTRIDE 388   // 384 + 4 pad: row-to-row bank offset of 4; 16B aligned rows

__global__ __launch_bounds__(256)
void k_gemm_wmma(const float* __restrict__ x,     // [B, C, N]
                 const float* __restrict__ mr,    // [B, C, N]
                 const float* __restrict__ W,     // [CO, 2C]
                 const float* __restrict__ bconv, // [CO]
                 float*       __restrict__ h)     // [B, CO, N]
{
  __shared__ float sW[16 * SWSTRIDE];

  const int b  = blockIdx.z;
  const int o0 = blockIdx.y * 16;

  // stage W[o0:o0+16, 0:384] into LDS as float4 (b128), coalesced
  for (int t = threadIdx.x; t < 16 * (CIN2 / 4); t += 256) {
    const int r = t / (CIN2 / 4), c4 = t - r * (CIN2 / 4);
    const float4 v = ((const float4*)(W + (o0 + r) * CIN2))[c4];
    *(float4*)&sW[r * SWSTRIDE + c4 * 4] = v;
  }
  __syncthreads();

  const int wave = threadIdx.x >> 5;
  const int lane = threadIdx.x & 31;
  const int lo   = lane & 15;
  const int hi   = lane >> 4;              // 0: K pair {0,1}, 1: K pair {2,3}
  const int n0   = (blockIdx.x * 8 + wave) * 16;

  const float* xb = x  + (size_t)b * CC * NN;
  const float* mb = mr + (size_t)b * CC * NN;

  v8f acc = {};
  for (int k0 = 0; k0 < CIN2; k0 += 4) {
    v2f a;
    a.x = sW[lo * SWSTRIDE + k0 + 2 * hi];
    a.y = sW[lo * SWSTRIDE + k0 + 2 * hi + 1];

    const int c = (k0 >> 1) + hi;          // input channel for this lane's K pair
    v2f bf;
    bf.x = xb[c * NN + n0 + lo];           // even c2 -> x
    bf.y = mb[c * NN + n0 + lo];           // odd  c2 -> mr

    acc = __builtin_amdgcn_wmma_f32_16x16x4_f32(
        /*neg_a=*/false, a, /*neg_b=*/false, bf,
        /*c_mod=*/(short)0, acc, /*reuse_a=*/false, /*reuse_b=*/false);
  }

  // D layout: VGPR r -> row (o0 + r + 8*hi), col (n0 + lo)
  float* hb = h + (size_t)b * CO * NN;
#pragma unroll
  for (int r = 0; r < 8; ++r) {
    const int o = o0 + r + 8 * hi;
    hb[o * NN + n0 + lo] = acc[r] + bconv[o];
  }
}

// ---------------------------------------------------------------------------
// Kernel 3: fused BatchNorm (batch statistics over B*N) + exact GELU.
// One block per output channel; deterministic LDS tree reduction; float4 I/O.
// ---------------------------------------------------------------------------
__global__ __launch_bounds__(512)
void k_bn_gelu(const float* __restrict__ h,      // [B, CO, N]
               const float* __restrict__ gamma,
               const float* __restrict__ beta,
               float*       __restrict__ out)    // [B, CO, N]
{
  __shared__ float red1[512];
  __shared__ float red2[512];
  const int o = blockIdx.x;

  float s1 = 0.f, s2 = 0.f;
  for (int b = 0; b < BB; ++b) {
    const float4* p = (const float4*)(h + ((size_t)b * CO + o) * NN);
    for (int q = threadIdx.x; q < NN / 4; q += 512) {
      const float4 v = p[q];
      s1 += v.x + v.y + v.z + v.w;
      s2 += v.x * v.x + v.y * v.y + v.z * v.z + v.w * v.w;
    }
  }
  red1[threadIdx.x] = s1;
  red2[threadIdx.x] = s2;
  __syncthreads();
  for (int s = 256; s > 0; s >>= 1) {
    if (threadIdx.x < s) {
      red1[threadIdx.x] += red1[threadIdx.x + s];
      red2[threadIdx.x] += red2[threadIdx.x + s];
    }
    __syncthreads();
  }
  const float inv  = 1.0f / (float)(BB * NN);
  const float mean = red1[0] * inv;
  const float var  = red2[0] * inv - mean * mean;
  const float rstd = rsqrtf(var + EPSBN);
  const float g = gamma[o], be = beta[o];
  const float is2 = 0.70710678118654752440f;   // 1/sqrt(2)

  for (int b = 0; b < BB; ++b) {
    const float4* p = (const float4*)(h   + ((size_t)b * CO + o) * NN);
    float4*       q = (float4*)      (out + ((size_t)b * CO + o) * NN);
    for (int t = threadIdx.x; t < NN / 4; t += 512) {
      float4 v = p[t];
      v.x = (v.x - mean) * rstd * g + be;
      v.y = (v.y - mean) * rstd * g + be;
      v.z = (v.z - mean) * rstd * g + be;
      v.w = (v.w - mean) * rstd * g + be;
      float4 r;
      r.x = 0.5f * v.x * (1.0f + erff(v.x * is2));
      r.y = 0.5f * v.y * (1.0f + erff(v.y * is2));
      r.z = 0.5f * v.z * (1.0f + erff(v.z * is2));
      r.w = 0.5f * v.w * (1.0f + erff(v.w * is2));
      q[t] = r;
    }
  }
}

// ---------------------------------------------------------------------------
extern "C" void kernel_launch(void* const* d_in, const int* in_sizes, int n_in,
                              void* d_out, int out_size, void* d_ws, size_t ws_size,
                              hipStream_t stream) {
  const float* x      = (const float*)d_in[0];   // [B, C, N, 1]
  const float* W      = (const float*)d_in[1];   // [CO, 2C]
  const float* bconv  = (const float*)d_in[2];   // [CO]
  const float* gamma  = (const float*)d_in[3];   // [CO]
  const float* beta   = (const float*)d_in[4];   // [CO]
  const int*   eidx   = (const int*)d_in[5];     // [2, B, N, K]

  // workspace partition
  float* mr = (float*)d_ws;                                   // B*C*N floats
  float* h  = mr + (size_t)BB * CC * NN;                      // B*CO*N floats

  // 1) gather + max-relative (double-buffered async rows: 128KB dynamic LDS)
  {
    dim3 grid(NN / TILE_N, CC / CGROUP, BB);                  // (64, 16, 2)
    size_t shmem = 2 * NN * sizeof(float);                    // 128KB
    k_gather_mr<<<grid, 256, shmem, stream>>>(x, eidx, mr);
  }
  // 2) WMMA GEMM (1x1 conv)
  {
    dim3 grid(NN / (16 * 8), CO / 16, BB);                    // (128, 12, 2)
    k_gemm_wmma<<<grid, 256, 0, stream>>>(x, mr, W, bconv, h);
  }
  // 3) fused BN + GELU
  {
    k_bn_gelu<<<CO, 512, 0, stream>>>(h, gamma, beta, (float*)d_out);
  }
}